// SENet_36532991819829
// MI455X (gfx1250) — compile-verified
//
#include <hip/hip_runtime.h>
#include <cstdint>
#include <cstddef>

// ---------------------------------------------------------------------------
// SENet-style segment gating, MI455X (gfx1250, wave32).
// Memory-bound: 752 MB of HBM traffic, ~32 us floor at 23.3 TB/s.
// Strategy: one-pass streaming. Each 64-thread block (2 waves) stages a
// 32x188 f32 tile in LDS (async global->LDS B128 when available), each wave
// computes its 16x12 segment means with V_WMMA_F32_16X16X4_F32 (47 K-steps),
// gates come from a tiny per-thread MLP, and the scaled tile is written out
// with float4 stores. x is touched in HBM exactly once.
// ---------------------------------------------------------------------------

#define TPB        64
#define TILE_ROWS  32
#define NF         188
#define TILE_F4    (TILE_ROWS * NF / 4)   // 1504 float4 chunks per tile

typedef float v2f __attribute__((ext_vector_type(2)));
typedef float v8f __attribute__((ext_vector_type(8)));
typedef int   v4i __attribute__((vector_size(16)));   // matches async-lds builtin param

#define HAVE_WMMA4 __has_builtin(__builtin_amdgcn_wmma_f32_16x16x4_f32)
#define HAVE_ASYNC __has_builtin(__builtin_amdgcn_global_load_async_to_lds_b128)

// Fixed segment structure; constexpr so fully-unrolled loops fold these to
// immediates (no per-lane table loads).
__host__ __device__ constexpr int seg_of(int k) {
  const int Bnd[13] = {0,20,26,47,67,73,94,114,120,141,161,167,188};
  int s = 0;
  for (int i = 1; i < 12; ++i) if (k >= Bnd[i]) s = i;
  return s;
}
__host__ __device__ constexpr float inv_of(int s) {
  const int Bnd[13] = {0,20,26,47,67,73,94,114,120,141,161,167,188};
  return 1.0f / (float)(Bnd[s + 1] - Bnd[s]);
}

__device__ __forceinline__ void gates_from_means(const float* m,
                                                 const float* __restrict__ W1,
                                                 const float* __restrict__ b1,
                                                 const float* __restrict__ W2,
                                                 const float* __restrict__ b2,
                                                 float* g) {
  float h[3];
#pragma unroll
  for (int i = 0; i < 3; ++i) {
    float a = b1[i];
#pragma unroll
    for (int j = 0; j < 12; ++j) a = fmaf(W1[i * 12 + j], m[j], a);
    h[i] = fmaxf(a, 0.0f);
  }
#pragma unroll
  for (int s = 0; s < 12; ++s) {
    float z = b2[s];
#pragma unroll
    for (int i = 0; i < 3; ++i) z = fmaf(W2[s * 3 + i], h[i], z);
    g[s] = 1.0f / (1.0f + __expf(-z));
  }
}

__global__ __launch_bounds__(TPB) void senet_tile_kernel(
    const float* __restrict__ x,
    const float* __restrict__ W1, const float* __restrict__ b1,
    const float* __restrict__ W2, const float* __restrict__ b2,
    float* __restrict__ out) {
  __shared__ float tile[TILE_ROWS * NF];       // 24064 B, contiguous copy of the tile
  __shared__ float meansLds[TILE_ROWS * 16];   // 2 KB, [row][segment] (cols 12..15 junk)

  const int t    = threadIdx.x;
  const int lane = t & 31;
  const int w    = t >> 5;        // wave id: 0 or 1 (owns rows 16w..16w+15)
  const int hi   = lane >> 4;     // half-wave select (WMMA K split)
  const int n    = lane & 15;     // WMMA M/N lane index
  const size_t tileRow0 = (size_t)blockIdx.x * TILE_ROWS;

  // ---- Phase 1: HBM -> LDS, one contiguous 24 KB stream of B128s ----------
  {
    const float4* gsrc = (const float4*)(x + tileRow0 * NF);
#if HAVE_ASYNC
    for (int i = t; i < TILE_F4; i += TPB) {
      __builtin_amdgcn_global_load_async_to_lds_b128(
          (v4i*)(gsrc + i),          // global source (cast drops const)
          (v4i*)&tile[i * 4],        // LDS destination
          0, 0);
    }
    asm volatile("s_wait_asynccnt 0" ::: "memory");
#else
    for (int i = t; i < TILE_F4; i += TPB) {
      float4 v = gsrc[i];
      *(float4*)&tile[i * 4] = v;
    }
#endif
  }
  __syncthreads();

  // ---- Phase 2: segment means via V_WMMA_F32_16X16X4_F32 ------------------
  // D(16x16) += A(16x4: x tile rows) * B(4x16: averaging matrix).
  // A layout: lanes 0-15 hold row M=lane, K={4T,4T+1}; lanes 16-31 K={4T+2,4T+3}.
  // B layout mirrors: lanes 0-15 hold K={4T,4T+1} for N=lane; lanes 16-31 K={+2,+3}.
#if HAVE_WMMA4
  {
    v8f acc = {0.f, 0.f, 0.f, 0.f, 0.f, 0.f, 0.f, 0.f};
    const int rr = (w << 4) + n;  // tile row this lane feeds into A
#pragma unroll
    for (int T = 0; T < 47; ++T) {
      const int k0 = 4 * T;
      // A fragment: conflict-free ds_load_b64 (row stride 188 dwords).
      const float2 av = *(const float2*)&tile[rr * NF + k0 + 2 * hi];
      v2f a; a.x = av.x; a.y = av.y;
      // B fragment: MEAN_MAT[k][n] = (seg_of(k)==n) ? 1/len : 0, k compile-time.
      const float bx0 = (n == seg_of(k0 + 0)) ? inv_of(seg_of(k0 + 0)) : 0.f;
      const float by0 = (n == seg_of(k0 + 1)) ? inv_of(seg_of(k0 + 1)) : 0.f;
      const float bx1 = (n == seg_of(k0 + 2)) ? inv_of(seg_of(k0 + 2)) : 0.f;
      const float by1 = (n == seg_of(k0 + 3)) ? inv_of(seg_of(k0 + 3)) : 0.f;
      v2f b; b.x = hi ? bx1 : bx0; b.y = hi ? by1 : by0;
      acc = __builtin_amdgcn_wmma_f32_16x16x4_f32(
          /*neg_a=*/false, a, /*neg_b=*/false, b,
          /*c_mod=*/(short)0, acc, /*reuse_a=*/false, /*reuse_b=*/false);
    }
    // C layout: VGPR j, lanes 0-15 -> M=j, N=lane; lanes 16-31 -> M=j+8.
#pragma unroll
    for (int j = 0; j < 8; ++j)
      meansLds[((w << 4) + j + (hi << 3)) * 16 + n] = acc[j];
  }
#else
  // Scalar fallback: thread t (<32) reduces row t from LDS.
  if (t < TILE_ROWS) {
    float s[12] = {0.f, 0.f, 0.f, 0.f, 0.f, 0.f, 0.f, 0.f, 0.f, 0.f, 0.f, 0.f};
#pragma unroll
    for (int k = 0; k < NF; ++k) s[seg_of(k)] += tile[t * NF + k];
#pragma unroll
    for (int i = 0; i < 12; ++i) meansLds[t * 16 + i] = s[i] * inv_of(i);
  }
#endif
  __syncthreads();

  // ---- Phase 3: per-row MLP -> gates, then scale & store ------------------
  const int rg = t & 31;  // wave 0 and wave 1 both cover rows 0..31 (halves of cols)
  float m[12];
#pragma unroll
  for (int i = 0; i < 12; ++i) m[i] = meansLds[rg * 16 + i];
  float g[12];
  gates_from_means(m, W1, b1, W2, b2, g);

  const float* xr = &tile[rg * NF];
  float* orow = out + (tileRow0 + (size_t)rg) * NF;
  if (t < 32) {  // wave 0: cols 0..95 (no divergence within a wave)
#pragma unroll
    for (int c = 0; c < 96; c += 4) {
      float4 v = *(const float4*)&xr[c];
      v.x *= g[seg_of(c + 0)]; v.y *= g[seg_of(c + 1)];
      v.z *= g[seg_of(c + 2)]; v.w *= g[seg_of(c + 3)];
      *(float4*)&orow[c] = v;
    }
  } else {       // wave 1: cols 96..187
#pragma unroll
    for (int c = 96; c < NF; c += 4) {
      float4 v = *(const float4*)&xr[c];
      v.x *= g[seg_of(c + 0)]; v.y *= g[seg_of(c + 1)];
      v.z *= g[seg_of(c + 2)]; v.w *= g[seg_of(c + 3)];
      *(float4*)&orow[c] = v;
    }
  }
}

// Generic tail for row counts not divisible by 32 (not used for B=500000).
__global__ void senet_tail_kernel(const float* __restrict__ x,
                                  const float* __restrict__ W1, const float* __restrict__ b1,
                                  const float* __restrict__ W2, const float* __restrict__ b2,
                                  float* __restrict__ out, int row0, int nrows) {
  int r = row0 + (int)(blockIdx.x * blockDim.x + threadIdx.x);
  if (r >= nrows) return;
  const float* xr = x + (size_t)r * NF;
  float s[12] = {0.f, 0.f, 0.f, 0.f, 0.f, 0.f, 0.f, 0.f, 0.f, 0.f, 0.f, 0.f};
#pragma unroll
  for (int k = 0; k < NF; ++k) s[seg_of(k)] += xr[k];
  float m[12];
#pragma unroll
  for (int i = 0; i < 12; ++i) m[i] = s[i] * inv_of(i);
  float g[12];
  gates_from_means(m, W1, b1, W2, b2, g);
  float* orow = out + (size_t)r * NF;
#pragma unroll
  for (int k = 0; k < NF; ++k) orow[k] = xr[k] * g[seg_of(k)];
}

extern "C" void kernel_launch(void* const* d_in, const int* in_sizes, int n_in,
                              void* d_out, int out_size, void* d_ws, size_t ws_size,
                              hipStream_t stream) {
  (void)n_in; (void)out_size; (void)d_ws; (void)ws_size;
  const float* x  = (const float*)d_in[0];
  const float* W1 = (const float*)d_in[1];
  const float* b1 = (const float*)d_in[2];
  const float* W2 = (const float*)d_in[3];
  const float* b2 = (const float*)d_in[4];
  float* out = (float*)d_out;

  const int rows  = in_sizes[0] / NF;          // 500000
  const int tiles = rows / TILE_ROWS;          // 15625 (exact)
  const int rem   = rows - tiles * TILE_ROWS;  // 0 for this problem

  if (tiles > 0)
    senet_tile_kernel<<<tiles, TPB, 0, stream>>>(x, W1, b1, W2, b2, out);
  if (rem > 0)
    senet_tail_kernel<<<(rem + 63) / 64, 64, 0, stream>>>(x, W1, b1, W2, b2, out,
                                                          tiles * TILE_ROWS, rows);
}